// double_exponential_filter_layer_33801392620110
// MI455X (gfx1250) — compile-verified
//
#include <hip/hip_runtime.h>

// ============================================================================
// Holt double-exponential smoothing (alpha=beta=0.5) over [B=32, N=2048, T=1024]
//
// Roofline: 512 MB total traffic @ 23.3 TB/s  =>  ~22 us floor. Memory bound.
// The recurrence v_t = M v_{t-1} + c x_t (M,c constant) is linearized into
// 16-step blocks:  S_blk = W (16x16 const) * X_blk + boundary(s_in, b_in),
// computed with 4x chained V_WMMA_F32_16X16X4_F32 per block (K=16 total).
// This breaks the 1024-long serial FMA chain into 64 cheap state hops and
// keeps everything in f32 on the matrix pipe.
//
// Quirky init handling: reference == full scan from (0,0) with x0,x1 zeroed,
// then out[1] := x[1]. We implement exactly that (segment-0 special case).
// ============================================================================

typedef __attribute__((ext_vector_type(2))) float v2f;
typedef __attribute__((ext_vector_type(8))) float v8f;

// Compile-time coefficient tables.
// M = [[0.5, 0.5], [-0.25, 0.75]], c = (0.5, 0.25)
//   w_k   = e1 . (M^k c)              -> W[j][i] = w_{j-i} (i<=j), Toeplitz
//   (Ps,Pb)[j] = e1 . M^{j+1}          (boundary contribution to output row j)
//   Bc[j] = weight of s_j in b_15:  b15 = 0.5 s15 - sum_{j<15} 0.5^{16-j} s_j
//                                        - 2^-16 s_in + 2^-16 b_in
struct ScanTables {
  float W[16][16];
  float Ps[16], Pb[16], Bc[16];
  float Cend;
  constexpr ScanTables() : W{}, Ps{}, Pb{}, Bc{}, Cend(0.0f) {
    double w[16] = {};
    double qs = 0.5, qb = 0.25;               // q_k = M^k c
    for (int k = 0; k < 16; ++k) {
      w[k] = qs;
      double ns = 0.5 * qs + 0.5 * qb;
      double nb = -0.25 * qs + 0.75 * qb;
      qs = ns; qb = nb;
    }
    for (int j = 0; j < 16; ++j)
      for (int i = 0; i < 16; ++i)
        W[j][i] = (i <= j) ? (float)w[j - i] : 0.0f;
    double rs = 1.0, rb = 0.0;                // r_k = e1 M^k, need k=1..16
    for (int j = 0; j < 16; ++j) {
      double nrs = 0.5 * rs - 0.25 * rb;
      double nrb = 0.5 * rs + 0.75 * rb;
      rs = nrs; rb = nrb;
      Ps[j] = (float)rs; Pb[j] = (float)rb;
    }
    double half[17] = {};
    half[0] = 1.0;
    for (int k = 1; k <= 16; ++k) half[k] = half[k - 1] * 0.5;
    for (int j = 0; j < 15; ++j) Bc[j] = (float)(-half[16 - j]);
    Bc[15] = 0.5f;
    Cend = (float)half[16];
  }
};

static __device__ __constant__ ScanTables g_tab{};

#define WAVES_PER_BLOCK 4
#define LDS_STRIDE 20  // floats; multiple of 4 -> 16B aligned b128, conflict-free

__global__ __launch_bounds__(128) void holt_scan_wmma_kernel(
    const float* __restrict__ x, float* __restrict__ out) {
  __shared__ float lds[WAVES_PER_BLOCK][16 * LDS_STRIDE];

  const int lane = threadIdx.x & 31;
  const int wave = threadIdx.x >> 5;
  const int n    = lane & 15;   // chain within this wave's 16-chain tile
  const int hi   = lane >> 4;   // lane-half selector (K/row split per ISA layout)

  const long baseChain = (long)blockIdx.x * (WAVES_PER_BLOCK * 16) + wave * 16;
  const float* __restrict__ xrow = x + (baseChain + n) * 1024;
  float* __restrict__ ldsw = &lds[wave][0];

  // A fragments: A[m][k], m = lane&15, VGPRv holds k = 4*s + v + 2*hi
  v2f aW[4];
#pragma unroll
  for (int s = 0; s < 4; ++s) {
    aW[s].x = g_tab.W[n][4 * s + 0 + 2 * hi];
    aW[s].y = g_tab.W[n][4 * s + 1 + 2 * hi];
  }
  // Per-lane boundary/reduction coefficients for this lane's 8 output rows
  float ps[8], pb[8], bc[8];
#pragma unroll
  for (int r = 0; r < 8; ++r) {
    const int j = r + 8 * hi;
    ps[r] = g_tab.Ps[j];
    pb[r] = g_tab.Pb[j];
    bc[r] = g_tab.Bc[j];
  }
  const float cend = g_tab.Cend;

  float s_in = 0.0f, b_in = 0.0f;

  for (int seg = 0; seg < 64; ++seg) {
    const int t0 = seg * 16;

    // Prefetch two segments ahead (next 128B line of each chain row)
    __builtin_prefetch(xrow + t0 + 32, 0, 1);

    // B fragments: X[k][chain], k = 4*s + v + 2*hi  -> one float2 per slice
    v2f bF[4];
#pragma unroll
    for (int s = 0; s < 4; ++s)
      bF[s] = *(const v2f*)(xrow + t0 + 4 * s + 2 * hi);

    // Segment 0: scan with x0 = x1 = 0 (these live in lo-lanes' slice-0 pair)
    const float x1_val = bF[0].y;  // x[chain][1] on lo lanes at seg 0
    if (seg == 0 && hi == 0) {
      bF[0].x = 0.0f;
      bF[0].y = 0.0f;
    }

    // C accumulator pre-filled with boundary terms:  Ps[j]*s_in + Pb[j]*b_in
    v8f acc;
#pragma unroll
    for (int r = 0; r < 8; ++r)
      acc[r] = fmaf(ps[r], s_in, pb[r] * b_in);

    // S_blk = W * X_blk + boundary : 4 chained f32 WMMAs (K = 4 each)
#pragma unroll
    for (int s = 0; s < 4; ++s)
      acc = __builtin_amdgcn_wmma_f32_16x16x4_f32(
          false, aW[s], false, bF[s], (short)0, acc, false, false);

    // ---- state propagation (must use true scan values, before out[1] fixup)
    // s_in' = s_15 (row 15 lives in acc[7] of hi lanes)
    const float s15 = __shfl(acc[7], 16 + n, 32);
    float part = 0.0f;
#pragma unroll
    for (int r = 0; r < 8; ++r) part = fmaf(bc[r], acc[r], part);
    part += __shfl_xor(part, 16, 32);  // combine j=0..7 and j=8..15 halves
    const float b_next = part + cend * (b_in - s_in);
    s_in = s15;
    b_in = b_next;

    // out[1] = x[1] fixup (display value only; state already computed)
    if (seg == 0 && hi == 0) acc[1] = x1_val;

    // ---- stage D to LDS [chain][t] (row j = r + 8*hi), then coalesced store
    const int cbase = n * LDS_STRIDE + 8 * hi;
    *(float4*)(ldsw + cbase + 0) =
        make_float4(acc[0], acc[1], acc[2], acc[3]);
    *(float4*)(ldsw + cbase + 4) =
        make_float4(acc[4], acc[5], acc[6], acc[7]);

    asm volatile("s_wait_dscnt 0" ::: "memory");

#pragma unroll
    for (int q = 0; q < 2; ++q) {
      const int flat = lane + 32 * q;  // 0..63
      const int ch   = flat >> 2;      // chain 0..15
      const int tq   = flat & 3;       // quad within 16 timesteps
      const float4 v = *(const float4*)(ldsw + ch * LDS_STRIDE + tq * 4);
      *(float4*)(out + (baseChain + ch) * 1024 + t0 + tq * 4) = v;
    }
    // LDS ops from the same wave complete in order -> next seg's stores are
    // safely ordered after this seg's loads.
  }
}

extern "C" void kernel_launch(void* const* d_in, const int* in_sizes, int n_in,
                              void* d_out, int out_size, void* d_ws, size_t ws_size,
                              hipStream_t stream) {
  (void)n_in; (void)out_size; (void)d_ws; (void)ws_size;
  const float* x = (const float*)d_in[0];
  float* out = (float*)d_out;

  const int T = 1024;
  const int chains = in_sizes[0] / T;                // 32 * 2048 = 65536
  const int chainsPerBlock = WAVES_PER_BLOCK * 16;   // 64
  const int blocks = chains / chainsPerBlock;        // 1024

  holt_scan_wmma_kernel<<<dim3(blocks), dim3(128), 0, stream>>>(x, out);
}